// Intersection_89567247991565
// MI455X (gfx1250) — compile-verified
//
#include <hip/hip_runtime.h>
#include <hip/hip_bf16.h>

// CDNA5 (gfx1250) WMMA vector types
typedef __attribute__((ext_vector_type(2))) float v2f;
typedef __attribute__((ext_vector_type(8))) float v8f;

// NT GEMM: C[M,N] = op( A[M,K] * B[N,K]^T ),  op = relu (RELU=true) or identity.
// Both A and B are row-major with K contiguous -> symmetric WMMA fragment loads.
//
// V_WMMA_F32_16X16X4_F32 fragment layout (wave32):
//   A (16x4):  lane L holds A[row = L&15][k = kk+2*(L>>4) + {0,1}] in v0,v1
//   B (4x16):  lane L holds B[k = kk+2*(L>>4) + {0,1}][col = L&15]  in v0,v1
//              and Bmath[k][n] == Browmajor[n][k]  -> identical addressing to A.
//   C/D (16x16, 8 VGPRs): VGPR r: lanes0-15 -> (M=r, N=lane), lanes16-31 -> (M=r+8).
//
// Wave tiling: each wave computes 32(M) x 64(N): 2 A-tiles x 4 B-tiles = 8
// accumulators, so each k-step is 6 fragment loads feeding 8 WMMAs -- raises
// FLOP per byte of L0 traffic, the limiting path for the K=4 f32 WMMA shape.
//
// Addressing: loads are expressed as uniform_base + 32-bit per-lane offset so
// the backend emits global_load (saddr + voffset + imm) -- the per-lane offset
// VGPR is loop-invariant and the k-advance is scalar, eliminating per-lane
// 64-bit pointer bumps (v_add_nc_u64) from the hot loop.
//
// Block = 256 threads = 8 waves arranged 4(M) x 2(N) -> 128 x 128 per block.
template <bool RELU>
__global__ __launch_bounds__(256) void gemm_nt_wmma_f32(
    const float* __restrict__ A,   // [Mtotal, K] row-major
    const float* __restrict__ B,   // [N, K] row-major (used transposed)
    float* __restrict__ C,         // [Mtotal, N] row-major
    int K, int N)
{
    const int lane = threadIdx.x & 31;
    const int wave = threadIdx.x >> 5;
    const int l16  = lane & 15;
    const int h    = lane >> 4;          // lane half selects k pair {0,1} vs {2,3}

    const int mBase = blockIdx.y * 128 + (wave >> 1) * 32;  // two 16-row tiles
    const int nBase = blockIdx.x * 128 + (wave & 1) * 64;   // four 16-col tiles

    // Loop-invariant per-lane element offsets (32-bit; max ~2.1M elements here).
    const unsigned offA = (unsigned)(mBase + l16) * (unsigned)K + 2u * h;
    const unsigned offB = (unsigned)(nBase + l16) * (unsigned)K + 2u * h;
    const unsigned tileOff = 16u * (unsigned)K;   // next 16-row/col tile

    v8f acc[2][4] = {};   // [mTile][nTile], register-resident (all loops unrolled)

    #pragma unroll 4
    for (unsigned kk = 0; kk < (unsigned)K; kk += 4) {
        v2f afrag[2];
        v2f bfrag[4];
        #pragma unroll
        for (int i = 0; i < 2; ++i)
            afrag[i] = *(const v2f*)(A + offA + (unsigned)i * tileOff + kk);
        #pragma unroll
        for (int j = 0; j < 4; ++j)
            bfrag[j] = *(const v2f*)(B + offB + (unsigned)j * tileOff + kk);

        #pragma unroll
        for (int i = 0; i < 2; ++i) {
            #pragma unroll
            for (int j = 0; j < 4; ++j) {
                // (neg_a, A, neg_b, B, c_mod, C, reuse_a, reuse_b)
                acc[i][j] = __builtin_amdgcn_wmma_f32_16x16x4_f32(
                    false, afrag[i], false, bfrag[j],
                    (short)0, acc[i][j], false, false);
            }
        }
    }

    // Store D (optionally relu-fused). C/D layout: VGPR r -> M = tileM + 8*h + r.
    #pragma unroll
    for (int i = 0; i < 2; ++i) {
        unsigned crow = (unsigned)(mBase + i * 16 + 8 * h) * (unsigned)N
                      + (unsigned)(nBase + l16);
        #pragma unroll
        for (int r = 0; r < 8; ++r) {
            #pragma unroll
            for (int j = 0; j < 4; ++j) {
                float e = acc[i][j][r];
                if (RELU) e = fmaxf(e, 0.0f);
                C[crow + 16u * j] = e;
            }
            crow += (unsigned)N;
        }
    }
}

extern "C" void kernel_launch(void* const* d_in, const int* in_sizes, int n_in,
                              void* d_out, int out_size, void* d_ws, size_t ws_size,
                              hipStream_t stream) {
    const float* a     = (const float*)d_in[0];   // [Na, K]
    const float* b     = (const float*)d_in[1];   // [Nb, K]
    const float* feats = (const float*)d_in[2];   // [F,  K]

    const int K  = 2048;
    const int Na = in_sizes[0] / K;   // 1024
    const int Nb = in_sizes[1] / K;   // 1024
    const int F  = in_sizes[2] / K;   // 128

    float* Sa = (float*)d_ws;                     // [Na, F] relu(a @ feats^T)
    float* Sb = Sa + (size_t)Na * F;              // [Nb, F] relu(b @ feats^T)
    float* out = (float*)d_out;                   // [Na, Nb]

    dim3 blk(256);

    // Stage 1: Sa = relu(a @ feats^T), Sb = relu(b @ feats^T)
    gemm_nt_wmma_f32<true><<<dim3(F / 128, Na / 128), blk, 0, stream>>>(
        a, feats, Sa, K, F);
    gemm_nt_wmma_f32<true><<<dim3(F / 128, Nb / 128), blk, 0, stream>>>(
        b, feats, Sb, K, F);

    // Stage 2: out = Sa @ Sb^T   (K = F = 128)
    gemm_nt_wmma_f32<false><<<dim3(Nb / 128, Na / 128), blk, 0, stream>>>(
        Sa, Sb, out, F, Nb);
}